// FeedForwardQuantum_65481071406396
// MI455X (gfx1250) — compile-verified
//
#include <hip/hip_runtime.h>
#include <hip/hip_bf16.h>

typedef __attribute__((ext_vector_type(2))) float v2f;
typedef __attribute__((ext_vector_type(8))) float v8f;

#define F_DIM 4096
#define E_DIM 1024
#define NQ    8
#define N_TOKENS 8192   // B*S = 4*2048

// ---------------- quantum circuit (2-level, 6 fixed gates) ----------------
__device__ __forceinline__ void g_rx(float t, float& ar, float& ai, float& br, float& bi) {
    float c = cosf(0.5f * t), s = sinf(0.5f * t);
    float n0r = c * ar + s * bi;
    float n0i = c * ai - s * br;
    float n1r = c * br + s * ai;
    float n1i = c * bi - s * ar;
    ar = n0r; ai = n0i; br = n1r; bi = n1i;
}
__device__ __forceinline__ void g_ry(float t, float& ar, float& ai, float& br, float& bi) {
    float c = cosf(0.5f * t), s = sinf(0.5f * t);
    float n0r = c * ar - s * br;
    float n0i = c * ai - s * bi;
    float n1r = s * ar + c * br;
    float n1i = s * ai + c * bi;
    ar = n0r; ai = n0i; br = n1r; bi = n1i;
}
__device__ __forceinline__ void g_rz(float t, float& ar, float& ai, float& br, float& bi) {
    float c = cosf(0.5f * t), s = sinf(0.5f * t);
    float n0r = c * ar + s * ai;       // e^{-it/2} * p0
    float n0i = c * ai - s * ar;
    float n1r = c * br - s * bi;       // e^{+it/2} * p1
    float n1i = c * bi + s * br;
    ar = n0r; ai = n0i; br = n1r; bi = n1i;
}

__device__ __forceinline__ float circuit_z(float th, float ph, float phase,
                                           float sq, float dr, float kr) {
    float p0r = 1.f, p0i = 0.f, p1r = 0.f, p1i = 0.f;
    g_rx(th,    p0r, p0i, p1r, p1i);
    g_ry(ph,    p0r, p0i, p1r, p1i);
    g_rz(phase, p0r, p0i, p1r, p1i);
    g_ry(sq,    p0r, p0i, p1r, p1i);
    g_rx(dr,    p0r, p0i, p1r, p1i);
    g_rz(kr,    p0r, p0i, p1r, p1i);
    return (p0r * p0r + p0i * p0i) - (p1r * p1r + p1i * p1i);
}

// ------------- kernel 1: h[f] = relu(w1[f,:] . z + b1[f]), f in [0,4096) -------------
__global__ void k_circuit_h(const float* __restrict__ bs_theta,
                            const float* __restrict__ bs_phi,
                            const float* __restrict__ phases,
                            const float* __restrict__ squeeze_r,
                            const float* __restrict__ disp_r,
                            const float* __restrict__ kerr,
                            const float* __restrict__ w1,
                            const float* __restrict__ b1,
                            float* __restrict__ h) {
    int f = blockIdx.x * blockDim.x + threadIdx.x;
    if (f >= F_DIM) return;
    float th = bs_theta[0], ph = bs_phi[0];
    float z0 = circuit_z(th, ph, phases[0], squeeze_r[0], disp_r[0], kerr[0]);
    float z1 = circuit_z(th, ph, phases[1], squeeze_r[1], disp_r[1], kerr[1]);
    const float* row = w1 + (size_t)f * NQ;
    float acc = row[0] * z0 + row[1] * z1;
    // wires 2..7 stay |0>: <Z> = 1
    acc += row[2] + row[3] + row[4] + row[5] + row[6] + row[7];
    acc += b1[f];
    h[f] = fmaxf(acc, 0.0f);
}

// ------------- kernel 2: out_vec[1024] = w2 @ h + b2, via fp32 WMMA -------------
// One block per 16-wide e-tile; 8 waves, each owns a 512-wide f-slice.
// WMMA GEMV trick: A[m,k] = h[f0+k] for all m (rows replicated),
//                  B[k,n] = w2[e0+n, f0+k]  ->  D[m,n] = partial out[e0+n] for every m.
__global__ void k_gemv_wmma(const float* __restrict__ w2,
                            const float* __restrict__ b2,
                            const float* __restrict__ h,
                            float* __restrict__ out_vec) {
    __shared__ float hs[F_DIM];         // 16 KB of the 320 KB LDS
    __shared__ float red[8 * 16];

    int tid = threadIdx.x;
    for (int i = tid; i < F_DIM; i += 256) hs[i] = h[i];
    __syncthreads();

    int wid   = tid >> 5;
    int lane  = tid & 31;
    int n     = lane & 15;              // output column within e-tile
    int khalf = (lane >> 4) << 1;       // 0 for lanes 0-15, 2 for lanes 16-31
    int e0    = blockIdx.x * 16;

    const float* wrow = w2 + (size_t)(e0 + n) * F_DIM + khalf;  // 8B aligned
    int f_begin = wid * (F_DIM / 8);
    int f_end   = f_begin + (F_DIM / 8);

    v8f acc = {0.f, 0.f, 0.f, 0.f, 0.f, 0.f, 0.f, 0.f};
    #pragma unroll 4
    for (int f0 = f_begin; f0 < f_end; f0 += 4) {
        v2f a, b;
        a.x = hs[f0 + khalf];
        a.y = hs[f0 + khalf + 1];
        float2 w = *(const float2*)(wrow + f0);   // global_load_b64
        b.x = w.x;
        b.y = w.y;
        // D = A(16x4) x B(4x16) + C ; emits v_wmma_f32_16x16x4_f32
        acc = __builtin_amdgcn_wmma_f32_16x16x4_f32(
            /*neg_a=*/false, a, /*neg_b=*/false, b,
            /*c_mod=*/(short)0, acc, /*reuse_a=*/false, /*reuse_b=*/false);
    }

    // D VGPR0: lanes 0-15 hold row M=0, lanes 16-31 row M=8 — all rows equal here.
    if (lane < 16) red[wid * 16 + n] = acc[0];
    __syncthreads();

    if (tid < 16) {
        float s = b2[e0 + tid];
        #pragma unroll
        for (int w = 0; w < 8; ++w) s += red[w * 16 + tid];
        out_vec[e0 + tid] = s;
    }
}

// ------------- kernel 3: broadcast out_vec[1024] to all 8192 token rows -------------
__global__ void k_bcast(const float* __restrict__ out_vec, float* __restrict__ out) {
    int tid = threadIdx.x;                         // 256 threads * float4 = 1024 floats/row
    float4 v = ((const float4*)out_vec)[tid];      // held in registers
    size_t row0 = (size_t)blockIdx.x * 32;
    float4* dst = (float4*)out + row0 * (E_DIM / 4) + tid;
    #pragma unroll
    for (int r = 0; r < 32; ++r)
        dst[(size_t)r * (E_DIM / 4)] = v;          // global_store_b128, streaming
}

extern "C" void kernel_launch(void* const* d_in, const int* in_sizes, int n_in,
                              void* d_out, int out_size, void* d_ws, size_t ws_size,
                              hipStream_t stream) {
    // setup_inputs order:
    // 0:x 1:bs_theta 2:bs_phi 3:phases 4:squeeze_r 5:disp_r 6:kerr 7:w1 8:b1 9:w2 10:b2
    const float* bs_theta  = (const float*)d_in[1];
    const float* bs_phi    = (const float*)d_in[2];
    const float* phases    = (const float*)d_in[3];
    const float* squeeze_r = (const float*)d_in[4];
    const float* disp_r    = (const float*)d_in[5];
    const float* kerr      = (const float*)d_in[6];
    const float* w1        = (const float*)d_in[7];
    const float* b1        = (const float*)d_in[8];
    const float* w2        = (const float*)d_in[9];
    const float* b2        = (const float*)d_in[10];
    float* out = (float*)d_out;

    float* h       = (float*)d_ws;       // 4096 floats
    float* out_vec = h + F_DIM;          // 1024 floats

    k_circuit_h<<<F_DIM / 256, 256, 0, stream>>>(bs_theta, bs_phi, phases, squeeze_r,
                                                 disp_r, kerr, w1, b1, h);
    k_gemv_wmma<<<E_DIM / 16, 256, 0, stream>>>(w2, b2, h, out_vec);
    k_bcast<<<N_TOKENS / 32, 256, 0, stream>>>(out_vec, out);
}